// PairwiseRelationModule_81415400063716
// MI455X (gfx1250) — compile-verified
//
#include <hip/hip_runtime.h>

#define BD  2
#define NN  256
#define DD  256
#define HID 256

typedef __attribute__((ext_vector_type(2))) float v2f;
typedef __attribute__((ext_vector_type(8))) float v8f;

// ---------------------------------------------------------------------------
// Kernel 1: u[b,n,h] = geom-projection of object n;  hlb[b,h] = lang@W1_l + b1
// ---------------------------------------------------------------------------
__global__ void prep_kernel(const float* __restrict__ lang,
                            const float* __restrict__ centers,
                            const float* __restrict__ sizes,
                            const float* __restrict__ W1,
                            const float* __restrict__ b1,
                            float* __restrict__ u,
                            float* __restrict__ hlb) {
  const int bn = blockIdx.x;           // b*NN + n
  const int h  = threadIdx.x;          // 0..255
  const int b  = bn >> 8;
  const int n  = bn & 255;

  const float c0 = centers[bn * 3 + 0] * 0.2f;
  const float c1 = centers[bn * 3 + 1] * 0.2f;
  const float c2 = centers[bn * 3 + 2] * 0.2f;
  const float s0 = sizes[bn * 3 + 0] * 0.5f;
  const float s1 = sizes[bn * 3 + 1] * 0.5f;
  const float s2 = sizes[bn * 3 + 2] * 0.5f;

  const float* Wg = W1 + (2 * DD) * HID;        // W1 rows 512..517
  float uv = c0 * Wg[0 * HID + h] + c1 * Wg[1 * HID + h] + c2 * Wg[2 * HID + h]
           + s0 * Wg[3 * HID + h] + s1 * Wg[4 * HID + h] + s2 * Wg[5 * HID + h];
  u[bn * HID + h] = uv;

  if (n == 0) {                                  // only 2 blocks do this
    const float* Wl = W1 + (2 * DD + 6) * HID;   // W1 rows 518..773
    float acc = b1[h];
    for (int d0 = 0; d0 < DD; ++d0)
      acc = fmaf(lang[b * DD + d0], Wl[d0 * HID + h], acc);
    hlb[b * HID + h] = acc;
  }
}

// ---------------------------------------------------------------------------
// Kernel 2 (WMMA fp32, 16x64 tile/wave, templated on P/Q half):
//   IS_Q == false : h_i = X @ W1_i ; P[b,n,h]  = h_i + u + hlb   (row-major)
//   IS_Q == true  : h_j = X @ W1_j ; QT[b,h,j] = h_j - u         (transposed)
// 128 waves per instantiation: 32 row-tiles x 4 col-groups of 64
// ---------------------------------------------------------------------------
template <bool IS_Q>
__global__ void hcomb_kernel(const float* __restrict__ X,
                             const float* __restrict__ W1,
                             const float* __restrict__ u,
                             const float* __restrict__ hlb,
                             float* __restrict__ dst) {   // P or QT
  const int lane  = threadIdx.x & 31;
  const int wid   = threadIdx.x >> 5;
  const int wv    = blockIdx.x * 8 + wid;   // 0..127
  const int m0    = (wv >> 2) << 4;         // 16-row tile (rows = b*NN+n)
  const int c0    = (wv & 3) << 6;          // 64-col group within this half
  const int l15   = lane & 15;
  const int khalf = lane >> 4;              // 0: K=k0,k0+1   1: K=k0+2,k0+3

  const float4* Arow = (const float4*)(X + (m0 + l15) * DD);
  const float*  Wb   = W1 + (IS_Q ? DD * HID : 0);
  const int     hc0  = c0 + l15;

  v8f acc[4] = {};
  for (int k0 = 0; k0 < DD; k0 += 4) {
    float4 av = Arow[k0 >> 2];
    v2f a;
    a.x = khalf ? av.z : av.x;
    a.y = khalf ? av.w : av.y;
    const int kb = k0 + (khalf << 1);
    const float* w0 = Wb + kb * HID + hc0;
#pragma unroll
    for (int c = 0; c < 4; ++c) {
      v2f bm;
      bm.x = w0[16 * c];
      bm.y = w0[HID + 16 * c];
      acc[c] = __builtin_amdgcn_wmma_f32_16x16x4_f32(false, a, false, bm,
                                                     (short)0, acc[c], false, false);
    }
  }

#pragma unroll
  for (int c = 0; c < 4; ++c) {
    const int hcol = hc0 + 16 * c;
#pragma unroll
    for (int r = 0; r < 8; ++r) {
      const int gr = m0 + r + (khalf << 3);   // global row = b*NN + n
      const int bb = gr >> 8;
      const float uv = u[gr * HID + hcol];
      const float dv = acc[c][r];
      if (IS_Q) {
        dst[(bb * HID + hcol) * NN + (gr & 255)] = dv - uv;     // QT[b,h,j]
      } else {
        dst[gr * HID + hcol] = dv + uv + hlb[bb * HID + hcol];  // P[b,n,h]
      }
    }
  }
}

// ---------------------------------------------------------------------------
// Kernel 3: scores[b,n,j] = W2 . relu(P[n]+Q[j]) + b2 ; masked softmax over j
// one block per (b, 4 rows of n); thread j amortizes each Q read over 4 rows
// ---------------------------------------------------------------------------
__global__ void scores_kernel(const float* __restrict__ P,
                              const float* __restrict__ QT,
                              const float* __restrict__ W2,
                              const float* __restrict__ b2,
                              const unsigned char* __restrict__ mask,
                              float* __restrict__ weights) {
  const int j  = threadIdx.x;
  const int b  = blockIdx.x >> 6;           // 64 blocks per batch
  const int n0 = (blockIdx.x & 63) << 2;    // 4 rows per block

  __shared__ float Pn[4][HID];
  __shared__ float W2s[HID];
  __shared__ float red[NN];

#pragma unroll
  for (int r = 0; r < 4; ++r)
    Pn[r][j] = P[((b << 8) + n0 + r) * HID + j];
  W2s[j] = W2[j];
  __syncthreads();

  const float* Qb = QT + b * HID * NN;
  float s[4] = {0.f, 0.f, 0.f, 0.f};
#pragma unroll 2
  for (int h = 0; h < HID; ++h) {
    const float q  = Qb[h * NN + j];        // coalesced: thread j -> column j
    const float wh = W2s[h];
#pragma unroll
    for (int r = 0; r < 4; ++r)
      s[r] = fmaf(wh, fmaxf(Pn[r][h] + q, 0.f), s[r]);
  }

  const float bias = b2[0];
  const bool  mj   = mask[(b << 8) + j];
#pragma unroll
  for (int r = 0; r < 4; ++r) {
    const bool valid = mj && mask[(b << 8) + n0 + r];
    s[r] = valid ? (s[r] + bias) : -1.0e9f;
  }

  // softmax over j, one row at a time
  for (int r = 0; r < 4; ++r) {
    red[j] = s[r];
    __syncthreads();
    for (int st = 128; st > 0; st >>= 1) {
      if (j < st) red[j] = fmaxf(red[j], red[j + st]);
      __syncthreads();
    }
    const float mx = red[0];
    __syncthreads();
    const float e = expf(s[r] - mx);
    red[j] = e;
    __syncthreads();
    for (int st = 128; st > 0; st >>= 1) {
      if (j < st) red[j] += red[j + st];
      __syncthreads();
    }
    weights[((b << 8) + n0 + r) * NN + j] = e / red[0];
    __syncthreads();                         // protect red[] before next row
  }
}

// ---------------------------------------------------------------------------
// Kernel 4 (WMMA fp32, 16x64 tile/wave):
//   enhanced[b] = feats[b] + weights[b](256x256) @ feats[b]
// ---------------------------------------------------------------------------
__global__ void context_kernel(const float* __restrict__ Wt,   // (B,N,N) weights
                               const float* __restrict__ F,    // (B,N,D) feats
                               float* __restrict__ out) {      // (B,N,D) enhanced
  const int lane  = threadIdx.x & 31;
  const int wid   = threadIdx.x >> 5;
  const int wg    = blockIdx.x * 8 + wid;   // 128 waves: 2 batches x 16 x 4
  const int b     = wg >> 6;
  const int t     = wg & 63;
  const int m0    = (t >> 2) << 4;          // 16-row tile
  const int h0    = (t & 3) << 6;           // 64-col group
  const int l15   = lane & 15;
  const int khalf = lane >> 4;

  const float4* Arow = (const float4*)(Wt + (b * NN + m0 + l15) * NN);
  const float*  Fb   = F + b * NN * DD;
  const int     hc0  = h0 + l15;

  v8f acc[4] = {};
  for (int k0 = 0; k0 < NN; k0 += 4) {
    float4 av = Arow[k0 >> 2];
    v2f a;
    a.x = khalf ? av.z : av.x;
    a.y = khalf ? av.w : av.y;
    const int kb = k0 + (khalf << 1);
    const float* f0 = Fb + kb * DD + hc0;
#pragma unroll
    for (int c = 0; c < 4; ++c) {
      v2f bm;
      bm.x = f0[16 * c];
      bm.y = f0[DD + 16 * c];
      acc[c] = __builtin_amdgcn_wmma_f32_16x16x4_f32(false, a, false, bm,
                                                     (short)0, acc[c], false, false);
    }
  }

#pragma unroll
  for (int c = 0; c < 4; ++c) {
    const int hc = hc0 + 16 * c;
#pragma unroll
    for (int r = 0; r < 8; ++r) {
      const int gn = m0 + r + (khalf << 3);
      out[(b * NN + gn) * DD + hc] = acc[c][r] + Fb[gn * DD + hc];
    }
  }
}

// ---------------------------------------------------------------------------
extern "C" void kernel_launch(void* const* d_in, const int* in_sizes, int n_in,
                              void* d_out, int out_size, void* d_ws, size_t ws_size,
                              hipStream_t stream) {
  (void)in_sizes; (void)n_in; (void)out_size; (void)ws_size;

  const float*         feats   = (const float*)d_in[0];        // (B,N,D)
  const float*         lang    = (const float*)d_in[1];        // (B,D)
  const float*         centers = (const float*)d_in[2];        // (B,N,3)
  const float*         sizes   = (const float*)d_in[3];        // (B,N,3)
  const unsigned char* mask    = (const unsigned char*)d_in[4];// (B,N) bool
  const float*         W1      = (const float*)d_in[5];        // (774,256)
  const float*         b1      = (const float*)d_in[6];        // (256)
  const float*         W2      = (const float*)d_in[7];        // (256)
  const float*         b2      = (const float*)d_in[8];        // scalar

  float* out      = (float*)d_out;
  float* enhanced = out;                     // B*N*D
  float* weights  = out + BD * NN * DD;      // B*N*N

  float* ws  = (float*)d_ws;
  float* u   = ws;                           // B*N*HID
  float* P   = ws + 1 * BD * NN * HID;       // B*N*HID
  float* QT  = ws + 2 * BD * NN * HID;       // B*HID*N (transposed)
  float* hlb = ws + 3 * BD * NN * HID;       // B*HID

  prep_kernel          <<<BD * NN, HID, 0, stream>>>(lang, centers, sizes, W1, b1, u, hlb);
  hcomb_kernel<false>  <<<16, 256, 0, stream>>>(feats, W1, u, hlb, P);
  hcomb_kernel<true>   <<<16, 256, 0, stream>>>(feats, W1, u, hlb, QT);
  scores_kernel        <<<BD * NN / 4, NN, 0, stream>>>(P, QT, W2, b2, mask, weights);
  context_kernel       <<<16, 256, 0, stream>>>(weights, feats, enhanced);
}